// HyperGCN_3453153706035
// MI455X (gfx1250) — compile-verified
//
#include <hip/hip_runtime.h>
#include <hip/hip_bf16.h>

#define N_NODES   500000
#define NUM_EDGES 527318
#define NNZ       4194304
#define F         64
#define NTILES    (N_NODES / 16)   // 31250 row-tiles of 16

typedef __attribute__((ext_vector_type(2))) float v2f;
typedef __attribute__((ext_vector_type(8))) float v8f;

// ---------------------------------------------------------------------------
// Pass 1: degrees. deg_v[n] += W_[e]; deg_e[e] += 1  for each incidence pair.
// ---------------------------------------------------------------------------
__global__ __launch_bounds__(256) void hgcn_deg_kernel(
    const int* __restrict__ node_idx, const int* __restrict__ edge_idx,
    const float* __restrict__ W_, float* __restrict__ deg_v,
    float* __restrict__ deg_e) {
  unsigned i = blockIdx.x * 256u + threadIdx.x;
  if (i >= NNZ) return;
  int n = node_idx[i];
  int e = edge_idx[i];
  atomicAdd(&deg_v[n], W_[e]);   // no-return -> global_atomic_add_f32
  atomicAdd(&deg_e[e], 1.0f);
}

// ---------------------------------------------------------------------------
// Pass 2: dvis = deg_v^-1/2 (in place); scale_e = W_/deg_e (in place).
// ---------------------------------------------------------------------------
__global__ __launch_bounds__(256) void hgcn_xform_kernel(
    float* __restrict__ deg_v, float* __restrict__ deg_e,
    const float* __restrict__ W_) {
  unsigned i = blockIdx.x * 256u + threadIdx.x;
  if (i < N_NODES)   deg_v[i] = rsqrtf(deg_v[i]);
  if (i < NUM_EDGES) deg_e[i] = W_[i] / deg_e[i];
}

// ---------------------------------------------------------------------------
// Pass 3: m[e,:] += dvis[n] * feats[n,:]   (one wave32 per nnz, float2/lane)
// ---------------------------------------------------------------------------
__global__ __launch_bounds__(256) void hgcn_scatter_edge_kernel(
    const int* __restrict__ node_idx, const int* __restrict__ edge_idx,
    const float* __restrict__ feats, const float* __restrict__ dvis,
    float* __restrict__ m) {
  unsigned tid  = blockIdx.x * 256u + threadIdx.x;
  unsigned i    = tid >> 5;          // nnz index (one wave per nnz)
  unsigned lane = tid & 31u;
  if (i >= NNZ) return;
  int n = node_idx[i];
  int e = edge_idx[i];
  float dv = dvis[n];
  float2 v = ((const float2*)(feats + (long)n * F))[lane];   // coalesced 256B/wave
  float* dst = m + (long)e * F + 2u * lane;
  atomicAdd(dst,     v.x * dv);
  atomicAdd(dst + 1, v.y * dv);
}

// ---------------------------------------------------------------------------
// Pass 4: ax[n,:] += scale_e[e] * m[e,:]   (accumulated directly into d_out)
// ---------------------------------------------------------------------------
__global__ __launch_bounds__(256) void hgcn_scatter_node_kernel(
    const int* __restrict__ node_idx, const int* __restrict__ edge_idx,
    const float* __restrict__ m, const float* __restrict__ scale_e,
    float* __restrict__ ax) {
  unsigned tid  = blockIdx.x * 256u + threadIdx.x;
  unsigned i    = tid >> 5;
  unsigned lane = tid & 31u;
  if (i >= NNZ) return;
  int n = node_idx[i];
  int e = edge_idx[i];
  float se = scale_e[e];
  float2 v = ((const float2*)(m + (long)e * F))[lane];
  float* dst = ax + (long)n * F + 2u * lane;
  atomicAdd(dst,     v.x * se);
  atomicAdd(dst + 1, v.y * se);
}

// ---------------------------------------------------------------------------
// Pass 5: out = sigmoid((dvis[row] * ax) @ lin_w + lin_b), in place on d_out.
// One wave per 16-row tile; V_WMMA_F32_16X16X4_F32, 4 N-tiles x 16 K-steps.
// A frag (16x4 f32): lane L -> M=L%16, K pair = 4k + 2*(L/16).
// B frag (4x16 f32): lane L -> N=L%16, same K pair.
// D (16x16 f32): VGPR r -> M=r (lanes 0-15), M=r+8 (lanes 16-31); N=L%16.
// Waves own disjoint 16-row blocks and read all A before writing -> in-place OK.
// ---------------------------------------------------------------------------
__global__ __launch_bounds__(256) void hgcn_gemm_sigmoid_kernel(
    float* __restrict__ axout, const float* __restrict__ dvis,
    const float* __restrict__ lin_w, const float* __restrict__ lin_b) {
  const int lane = threadIdx.x & 31;
  const int wave = threadIdx.x >> 5;
  const int tile = blockIdx.x * 8 + wave;
  if (tile >= NTILES) return;        // uniform per-wave exit; EXEC stays full

  const int r0   = tile * 16;
  const int mrow = lane & 15;        // M (for A) / N column (for B,D)
  const int half = lane >> 4;        // selects K sub-pair / upper M rows of D

  const float dv = dvis[r0 + mrow];
  const float* __restrict__ arow = axout + (long)(r0 + mrow) * F;

  v8f acc[4] = {v8f{}, v8f{}, v8f{}, v8f{}};

  for (int k = 0; k < 16; ++k) {     // K in chunks of 4
    const int ks = 4 * k + 2 * half;
    v2f a;
    a.x = arow[ks]     * dv;
    a.y = arow[ks + 1] * dv;
    const float* __restrict__ b0 = lin_w + (long)ks * F;        // row ks
    const float* __restrict__ b1 = lin_w + (long)(ks + 1) * F;  // row ks+1
#pragma unroll
    for (int t = 0; t < 4; ++t) {
      const int col = 16 * t + mrow;
      v2f b;
      b.x = b0[col];
      b.y = b1[col];
      acc[t] = __builtin_amdgcn_wmma_f32_16x16x4_f32(
          /*neg_a=*/false, a, /*neg_b=*/false, b,
          /*c_mod=*/(short)0, acc[t], /*reuse_a=*/false, /*reuse_b=*/false);
    }
  }

#pragma unroll
  for (int t = 0; t < 4; ++t) {
    const int col = 16 * t + mrow;
    const float bias = lin_b[col];
#pragma unroll
    for (int r = 0; r < 8; ++r) {
      const int row = r0 + r + 8 * half;
      float x = acc[t][r] + bias;
      axout[(long)row * F + col] = 1.0f / (1.0f + __expf(-x));
    }
  }
}

// ---------------------------------------------------------------------------
extern "C" void kernel_launch(void* const* d_in, const int* in_sizes, int n_in,
                              void* d_out, int out_size, void* d_ws, size_t ws_size,
                              hipStream_t stream) {
  const int*   node_idx = (const int*)d_in[0];
  const int*   edge_idx = (const int*)d_in[1];
  const float* feats    = (const float*)d_in[2];
  const float* W_       = (const float*)d_in[3];
  const float* lin_w    = (const float*)d_in[4];
  const float* lin_b    = (const float*)d_in[5];
  float* out = (float*)d_out;

  // Workspace layout (256B aligned): deg_v | deg_e(->scale_e) | m
  char* ws = (char*)d_ws;
  float* deg_v = (float*)ws;  ws += (((size_t)N_NODES   * 4 + 255) / 256) * 256;
  float* deg_e = (float*)ws;  ws += (((size_t)NUM_EDGES * 4 + 255) / 256) * 256;
  float* m     = (float*)ws;  // NUM_EDGES * 64 floats (135 MB)

  hipMemsetAsync(deg_v, 0, (size_t)N_NODES   * sizeof(float), stream);
  hipMemsetAsync(deg_e, 0, (size_t)NUM_EDGES * sizeof(float), stream);
  hipMemsetAsync(m,     0, (size_t)NUM_EDGES * F * sizeof(float), stream);
  hipMemsetAsync(out,   0, (size_t)N_NODES   * F * sizeof(float), stream);

  hgcn_deg_kernel<<<(NNZ + 255) / 256, 256, 0, stream>>>(
      node_idx, edge_idx, W_, deg_v, deg_e);

  hgcn_xform_kernel<<<(NUM_EDGES + 255) / 256, 256, 0, stream>>>(
      deg_v, deg_e, W_);

  // one wave32 per nnz -> NNZ*32 threads / 256 = NNZ/8 blocks
  hgcn_scatter_edge_kernel<<<NNZ / 8, 256, 0, stream>>>(
      node_idx, edge_idx, feats, deg_v, m);

  hgcn_scatter_node_kernel<<<NNZ / 8, 256, 0, stream>>>(
      node_idx, edge_idx, m, deg_e, out);

  hgcn_gemm_sigmoid_kernel<<<(NTILES + 7) / 8, 256, 0, stream>>>(
      out, deg_v, lin_w, lin_b);
}